// FactorTopy_73469710565991
// MI455X (gfx1250) — compile-verified
//
#include <hip/hip_runtime.h>
#include <math.h>
#include <stdint.h>

typedef float v2f __attribute__((ext_vector_type(2)));
typedef float v4f __attribute__((ext_vector_type(4)));
typedef float v8f __attribute__((ext_vector_type(8)));
typedef unsigned int u32x4 __attribute__((ext_vector_type(4)));
typedef int i32x4 __attribute__((ext_vector_type(4)));
typedef int i32x8 __attribute__((ext_vector_type(8)));

#define NPTS 30000
#define LL 12
#define BB 4
#define CC 768
#define DD 128
#define HH 16
#define WW 16
#define PE_DIM 192
#define PE_STRIDE 194   // pad to kill 16-way LDS bank conflicts
#define H_STRIDE 132
#define IMG_STRIDE 260  // 256 + 4 pad dwords (TDM pad_interval=256, pad_amount=4)
#define NBLK_SEL (NPTS / 16)   // 1875

#ifndef __has_builtin
#define __has_builtin(x) 0
#endif

#if defined(__HIP_DEVICE_COMPILE__) && __has_builtin(__builtin_amdgcn_tensor_load_to_lds)
#define HAVE_TDM 1
#else
#define HAVE_TDM 0
#endif

__device__ __forceinline__ v8f wmma_f32(v2f a, v2f b, v8f c) {
  // D = A(16x4 f32) * B(4x16 f32) + C(16x16 f32)
  return __builtin_amdgcn_wmma_f32_16x16x4_f32(false, a, false, b, (short)0, c,
                                               false, false);
}

__device__ __forceinline__ float gelu_exact(float x) {
  return 0.5f * x * (1.0f + erff(x * 0.70710678118654752f));
}

__device__ __forceinline__ float dot4(v4f a, v4f b) {
  return a.x * b.x + a.y * b.y + a.z * b.z + a.w * b.w;
}

// ---------------------------------------------------------------------------
// K0: gb[l,b,d] = sum_c global_tokens[l,b,c] * glob_w[l,d,c]
// ---------------------------------------------------------------------------
__global__ __launch_bounds__(256) void gb_kernel(
    const float* __restrict__ gt, const float* __restrict__ gw,
    float* __restrict__ gb) {
  int idx = blockIdx.x * 256 + threadIdx.x;
  if (idx >= LL * BB * DD) return;
  int d = idx % DD;
  int b = (idx / DD) % BB;
  int l = idx / (DD * BB);
  const float* gtp = gt + (l * BB + b) * CC;
  const float* gwp = gw + (l * DD + d) * CC;
  float s = 0.f;
  for (int c = 0; c < CC; ++c) s += gtp[c] * gwp[c];
  gb[idx] = s;
}

// ---------------------------------------------------------------------------
// K1: selector MLPs (WMMA f32 16x16x4).  96 threads = 3 waves, 16 points/WG.
// Wave i runs selector i (0=space, 1=layer, 2=scale) on a shared pe tile.
// ---------------------------------------------------------------------------
__global__ __launch_bounds__(96) void selector_kernel(
    const float* __restrict__ coords,
    const float* __restrict__ W1all, const float* __restrict__ b1all,
    const float* __restrict__ W2all, const float* __restrict__ b2all,
    const float* __restrict__ spaceW3, const float* __restrict__ layerW3,
    const float* __restrict__ scaleW3,
    float* __restrict__ sx, float* __restrict__ sy,
    float* __restrict__ slayer, float* __restrict__ sscale,
    float* __restrict__ entp, float* __restrict__ sqp) {
  __shared__ float spe[16 * PE_STRIDE];
  __shared__ float shb[3 * 16 * H_STRIDE];

  int tid = threadIdx.x;
  int wv = tid >> 5;
  int lane = tid & 31;
  int nbase = blockIdx.x * 16;
  int mrow = lane & 15;
  int hf = lane >> 4;

  // cooperative positional encoding: pe[m][c], c = j*64 + r, r<32:sin, else cos
  for (int idx = tid; idx < 16 * PE_DIM; idx += 96) {
    int m = idx / PE_DIM, c = idx % PE_DIM;
    int j = c >> 6, r = c & 63, k = r & 31;
    float f = 3.14159265358979323846f * exp2f((float)k * (8.0f / 31.0f));
    float ang = coords[(nbase + m) * 3 + j] * f;
    spe[m * PE_STRIDE + c] = (r < 32) ? sinf(ang) : cosf(ang);
  }
  __syncthreads();

  const float* W1 = W1all + wv * PE_DIM * DD;
  const float* W2 = W2all + wv * DD * DD;
  const float* b1 = b1all + wv * DD;
  const float* b2 = b2all + wv * DD;
  float* hb = shb + wv * 16 * H_STRIDE;

  __builtin_prefetch(W2, 0, 1);

  v8f acc[8];
#pragma unroll
  for (int t = 0; t < 8; ++t) acc[t] = (v8f){0.f,0.f,0.f,0.f,0.f,0.f,0.f,0.f};

  // Layer 1: [16 x 192] @ [192 x 128]
  for (int ks = 0; ks < PE_DIM / 4; ++ks) {
    int kk = 4 * ks + 2 * hf;
    v2f a = *(const v2f*)(spe + mrow * PE_STRIDE + kk);
#pragma unroll
    for (int t = 0; t < 8; ++t) {
      int col = 16 * t + mrow;
      v2f bf = {W1[(kk + 0) * DD + col], W1[(kk + 1) * DD + col]};
      acc[t] = wmma_f32(a, bf, acc[t]);
    }
  }
  // bias + exact GELU -> hb (A-operand staging for layer 2)
#pragma unroll
  for (int t = 0; t < 8; ++t) {
    int col = 16 * t + mrow;
    float bv = b1[col];
#pragma unroll
    for (int r = 0; r < 8; ++r) {
      int m = r + 8 * hf;
      hb[m * H_STRIDE + col] = gelu_exact(acc[t][r] + bv);
    }
    acc[t] = (v8f){0.f,0.f,0.f,0.f,0.f,0.f,0.f,0.f};
  }

  // Layer 2: [16 x 128] @ [128 x 128]
  for (int ks = 0; ks < DD / 4; ++ks) {
    int kk = 4 * ks + 2 * hf;
    v2f a = *(const v2f*)(hb + mrow * H_STRIDE + kk);
#pragma unroll
    for (int t = 0; t < 8; ++t) {
      int col = 16 * t + mrow;
      v2f bf = {W2[(kk + 0) * DD + col], W2[(kk + 1) * DD + col]};
      acc[t] = wmma_f32(a, bf, acc[t]);
    }
  }
  // bias + GELU -> hb = h2
#pragma unroll
  for (int t = 0; t < 8; ++t) {
    int col = 16 * t + mrow;
    float bv = b2[col];
#pragma unroll
    for (int r = 0; r < 8; ++r) {
      int m = r + 8 * hf;
      hb[m * H_STRIDE + col] = gelu_exact(acc[t][r] + bv);
    }
  }

  // heads (per-wave, tiny GEMVs from LDS)
  if (wv == 0) {
    // sel_space: 2 columns; lanes 0-15 -> x, 16-31 -> y; tanh then pixel coords
    const float* h = hb + mrow * H_STRIDE;
    float s = 0.f;
    for (int d = 0; d < DD; ++d) s += h[d] * spaceW3[d * 2 + hf];
    float t = tanhf(s);
    float px = (t + 1.0f) * 8.0f - 0.5f;  // (g+1)*W/2 - 0.5, W=H=16
    if (hf == 0) sx[nbase + mrow] = px; else sy[nbase + mrow] = px;
  } else if (wv == 1) {
    // sel_layer: 12 columns + softmax + entropy partial
    float ent = 0.f;
    if (lane < 16) {
      const float* h = hb + lane * H_STRIDE;
      float lg[12];
      float mx = -1e30f;
      for (int j = 0; j < 12; ++j) {
        float s = 0.f;
        for (int d = 0; d < DD; ++d) s += h[d] * layerW3[d * 12 + j];
        lg[j] = s;
        mx = fmaxf(mx, s);
      }
      float sum = 0.f;
      for (int j = 0; j < 12; ++j) { lg[j] = expf(lg[j] - mx); sum += lg[j]; }
      float inv = 1.0f / sum;
      for (int j = 0; j < 12; ++j) {
        float p = lg[j] * inv;
        slayer[(nbase + lane) * 12 + j] = p;
        ent += p * logf(p + 1e-8f);
      }
    }
    for (int off = 16; off; off >>= 1) ent += __shfl_xor(ent, off, 32);
    if (lane == 0) entp[blockIdx.x] = ent;
  } else {
    // sel_scale: 1 column + sigmoid + (s-0.5)^2 partial
    float sq = 0.f;
    if (lane < 16) {
      const float* h = hb + lane * H_STRIDE;
      float s = 0.f;
      for (int d = 0; d < DD; ++d) s += h[d] * scaleW3[d];
      float sg = 1.0f / (1.0f + expf(-s));
      sscale[nbase + lane] = sg;
      sq = (sg - 0.5f) * (sg - 0.5f);
    }
    for (int off = 16; off; off >>= 1) sq += __shfl_xor(sq, off, 32);
    if (lane == 0) sqp[blockIdx.x] = sq;
  }
}

// ---------------------------------------------------------------------------
// K2: feat[l,b,hw,d] = sum_c conv_w[l,d,c] * img[l,b,c,hw]
// One WG (8 waves) per (l,b); GEMM 128x256x768 via WMMA f32 16x16x4.
// B operand (img K-chunk, 16x256 f32) staged in LDS via the Tensor Data
// Mover: one wave issues TENSOR_LOAD_TO_LDS with a 2D D# whose pad feature
// (interval=256 dwords, amount=4 dwords) produces the 260-dword
// anti-bank-conflict row stride directly; waits on TENSORcnt, then the
// workgroup barrier publishes the tile to all 8 waves.
// ---------------------------------------------------------------------------
__global__ __launch_bounds__(256) void feat_kernel(
    const float* __restrict__ img_all, const float* __restrict__ cw_all,
    float* __restrict__ feat) {
  __shared__ float simg[16 * IMG_STRIDE];
  int lb = blockIdx.x;
  int l = lb >> 2;
  int tid = threadIdx.x;
  int wv = tid >> 5, lane = tid & 31;
  int mrow = lane & 15, hf = lane >> 4;

  const float* img = img_all + (size_t)lb * CC * (HH * WW);
  const float* cw = cw_all + (size_t)l * DD * CC + (size_t)(16 * wv + mrow) * CC;

  v8f acc[16];
#pragma unroll
  for (int t = 0; t < 16; ++t) acc[t] = (v8f){0.f,0.f,0.f,0.f,0.f,0.f,0.f,0.f};

#if !HAVE_TDM
  int srow = tid >> 4;          // 0..15 : K-row within chunk
  int scg = (tid & 15) * 16;    // 16-col group
#else
  // LDS byte offset of simg = low 32 bits of its flat (shared-aperture) addr
  unsigned lds_off = (unsigned)(uintptr_t)(&simg[0]);
#endif

  for (int kc = 0; kc < CC; kc += 16) {
    __syncthreads();
#if HAVE_TDM
    if (tid < 32) {  // one wave issues the DMA (EXEC ignored by TDM)
      unsigned long long ga =
          (unsigned long long)(uintptr_t)(img + (size_t)kc * (HH * WW));
      u32x4 g0;
      g0[0] = 1u;                               // count=1 (valid user D#)
      g0[1] = lds_off;                          // lds_addr (bytes)
      g0[2] = (unsigned)(ga & 0xffffffffull);   // global_addr[31:0]
      g0[3] = (unsigned)((ga >> 32) & 0x01ffffffull) | (2u << 30);  // type=2
      i32x8 g1;
      // data_size=4B (2), pad_enable, pad_interval=256 dwords (7),
      // pad_amount=4 dwords (3); workgroup_mask=0 (not in a cluster)
      g1[0] = (2 << 16) | (1 << 20) | (7 << 22) | (3 << 25);
      g1[1] = (256 << 16);        // tensor_dim0 = 256 (bits 63:48)
      g1[2] = (CC << 16);         // tensor_dim0 hi=0 | tensor_dim1 = 768
      g1[3] = (256 << 16);        // tensor_dim1 hi=0 | tile_dim0 = 256
      g1[4] = 16;                 // tile_dim1 = 16, tile_dim2 = 0 (2D)
      g1[5] = 256;                // tensor_dim0_stride = 256 elements
      g1[6] = 0;                  // dim0_stride hi | tensor_dim1_stride lo
      g1[7] = 0;                  // tensor_dim1_stride hi
      i32x4 g2 = {0, 0, 0, 0};    // 2D: dims 2/3 unused
      i32x4 g3 = {0, 0, 0, 0};
#if __clang_major__ >= 23
      i32x8 g45 = {0, 0, 0, 0, 0, 0, 0, 0};
      __builtin_amdgcn_tensor_load_to_lds(g0, g1, g2, g3, g45, 0);
#else
      __builtin_amdgcn_tensor_load_to_lds(g0, g1, g2, g3, 0);
#endif
#if __has_builtin(__builtin_amdgcn_s_wait_tensorcnt)
      __builtin_amdgcn_s_wait_tensorcnt(0);
#else
      asm volatile("s_wait_tensorcnt 0x0" ::: "memory");
#endif
    }
#else
    {  // cooperative stage of 16 K-rows x 256 cols (fallback path)
      const float* src = img + (size_t)(kc + srow) * (HH * WW) + scg;
      float* dst = simg + srow * IMG_STRIDE + scg;
#pragma unroll
      for (int q = 0; q < 4; ++q)
        *(v4f*)(dst + 4 * q) = *(const v4f*)(src + 4 * q);
      if (kc + 16 < CC) __builtin_prefetch(src + 16 * (HH * WW), 0, 1);
    }
#endif
    __syncthreads();
#pragma unroll
    for (int ks = 0; ks < 4; ++ks) {
      int rloc = 4 * ks + 2 * hf;
      v2f a = *(const v2f*)(cw + kc + rloc);
#pragma unroll
      for (int t = 0; t < 16; ++t) {
        v2f bf = {simg[rloc * IMG_STRIDE + 16 * t + mrow],
                  simg[(rloc + 1) * IMG_STRIDE + 16 * t + mrow]};
        acc[t] = wmma_f32(a, bf, acc[t]);
      }
    }
  }

  // store transposed: feat[lb][hw][d], d contiguous (512B rows for gathers)
#pragma unroll
  for (int t = 0; t < 16; ++t) {
    int hw = 16 * t + mrow;
    int dbase = 16 * wv + 8 * hf;   // VGPR r holds d = dbase + r
    float* dst = feat + ((size_t)lb * (HH * WW) + hw) * DD + dbase;
    v4f lo = __builtin_shufflevector(acc[t], acc[t], 0, 1, 2, 3);
    v4f hi = __builtin_shufflevector(acc[t], acc[t], 4, 5, 6, 7);
    *(v4f*)dst = lo;
    *(v4f*)(dst + 4) = hi;
  }
}

// ---------------------------------------------------------------------------
// K3: per-point bilinear gather + layer mix + folded D-reduction.
// One wave per point; lanes split D into 4-float chunks.
// ---------------------------------------------------------------------------
__global__ __launch_bounds__(256) void sample_kernel(
    const float* __restrict__ feat, const float* __restrict__ gb,
    const float* __restrict__ sx, const float* __restrict__ sy,
    const float* __restrict__ slayer, const float* __restrict__ sscale,
    const float* __restrict__ wout, const float* __restrict__ bias,
    float* __restrict__ y) {
  int n = blockIdx.x * 8 + (threadIdx.x >> 5);
  int lane = threadIdx.x & 31;
  if (n >= NPTS) return;

  float px = sx[n], py = sy[n];
  float x0f = floorf(px), y0f = floorf(py);
  int x0 = (int)x0f, y0 = (int)y0f;
  int x1 = x0 + 1, y1 = y0 + 1;
  float wx1 = px - x0f, wy1 = py - y0f;
  float wx0 = 1.f - wx1, wy0 = 1.f - wy1;
  float vx0 = (x0 >= 0 && x0 < WW) ? 1.f : 0.f;
  float vx1 = (x1 >= 0 && x1 < WW) ? 1.f : 0.f;
  float vy0 = (y0 >= 0 && y0 < HH) ? 1.f : 0.f;
  float vy1 = (y1 >= 0 && y1 < HH) ? 1.f : 0.f;
  float a00 = wx0 * wy0 * vx0 * vy0;
  float a10 = wx1 * wy0 * vx1 * vy0;
  float a01 = wx0 * wy1 * vx0 * vy1;
  float a11 = wx1 * wy1 * vx1 * vy1;
  int x0c = min(max(x0, 0), WW - 1), x1c = min(max(x1, 0), WW - 1);
  int y0c = min(max(y0, 0), HH - 1), y1c = min(max(y1, 0), HH - 1);
  int dl = lane * 4;
  int o00 = (y0c * WW + x0c) * DD + dl;
  int o10 = (y0c * WW + x1c) * DD + dl;
  int o01 = (y1c * WW + x0c) * DD + dl;
  int o11 = (y1c * WW + x1c) * DD + dl;

  v4f wo = *(const v4f*)(wout + (size_t)n * DD + dl);
  float accL[BB] = {0.f, 0.f, 0.f, 0.f};
  v4f gv[BB];
#pragma unroll
  for (int b = 0; b < BB; ++b) gv[b] = (v4f){0.f, 0.f, 0.f, 0.f};

  for (int l = 0; l < LL; ++l) {
    float wl = slayer[n * LL + l];
#pragma unroll
    for (int b = 0; b < BB; ++b) {
      const float* fb = feat + (size_t)(l * BB + b) * (HH * WW) * DD;
      v4f v = a00 * (*(const v4f*)(fb + o00)) + a10 * (*(const v4f*)(fb + o10))
            + a01 * (*(const v4f*)(fb + o01)) + a11 * (*(const v4f*)(fb + o11));
      accL[b] += wl * dot4(v, wo);
      gv[b] += wl * (*(const v4f*)(gb + (l * BB + b) * DD + dl));
    }
  }

  float s = sscale[n];
  float os = 1.f - s;
#pragma unroll
  for (int b = 0; b < BB; ++b) {
    float tot = os * accL[b] + s * dot4(gv[b], wo);
    for (int off = 16; off; off >>= 1) tot += __shfl_xor(tot, off, 32);
    if (lane == 0) y[(size_t)b * NPTS + n] = tot * (1.0f / DD) + bias[n];
  }
}

// ---------------------------------------------------------------------------
// K4: deterministic serial reduction of reg partials
// ---------------------------------------------------------------------------
__global__ void finalize_kernel(const float* __restrict__ entp,
                                const float* __restrict__ sqp,
                                float* __restrict__ reg_out) {
  if (threadIdx.x == 0 && blockIdx.x == 0) {
    float ent = 0.f, sq = 0.f;
    for (int i = 0; i < NBLK_SEL; ++i) { ent += entp[i]; sq += sqp[i]; }
    float ent_mean = ent / (float)NPTS;
    float sq_mean = sq / (float)NPTS;
    reg_out[0] = ent_mean / logf(12.0f) + sq_mean;
  }
}

// ---------------------------------------------------------------------------
extern "C" void kernel_launch(void* const* d_in, const int* in_sizes, int n_in,
                              void* d_out, int out_size, void* d_ws,
                              size_t ws_size, hipStream_t stream) {
  const float* local_tokens  = (const float*)d_in[0];
  const float* global_tokens = (const float*)d_in[1];
  const float* coords        = (const float*)d_in[2];
  const float* conv_w        = (const float*)d_in[3];
  const float* glob_w        = (const float*)d_in[4];
  const float* sel_W1        = (const float*)d_in[5];
  const float* sel_b1        = (const float*)d_in[6];
  const float* sel_W2        = (const float*)d_in[7];
  const float* sel_b2        = (const float*)d_in[8];
  const float* space_W3      = (const float*)d_in[9];
  const float* layer_W3      = (const float*)d_in[10];
  const float* scale_W3      = (const float*)d_in[11];
  const float* out_weight    = (const float*)d_in[12];
  const float* out_bias      = (const float*)d_in[13];
  float* y = (float*)d_out;

  float* ws = (float*)d_ws;
  float* entp = ws;                       // 1875 (pad 2048)
  float* sqp  = ws + 2048;                // 1875 (pad 2048)
  float* gb   = ws + 4096;                // 12*4*128 = 6144
  float* sx   = gb + LL * BB * DD;        // 30000
  float* sy   = sx + NPTS;                // 30000
  float* sl   = sy + NPTS;                // 360000
  float* ss   = sl + NPTS * LL;           // 30000
  float* feat = ss + NPTS;                // 12*4*256*128 = 1,572,864 (~6.3MB)

  hipLaunchKernelGGL(gb_kernel, dim3((LL * BB * DD + 255) / 256), dim3(256), 0,
                     stream, global_tokens, glob_w, gb);
  hipLaunchKernelGGL(selector_kernel, dim3(NBLK_SEL), dim3(96), 0, stream,
                     coords, sel_W1, sel_b1, sel_W2, sel_b2, space_W3,
                     layer_W3, scale_W3, sx, sy, sl, ss, entp, sqp);
  hipLaunchKernelGGL(feat_kernel, dim3(LL * BB), dim3(256), 0, stream,
                     local_tokens, conv_w, feat);
  hipLaunchKernelGGL(sample_kernel, dim3(NPTS / 8), dim3(256), 0, stream, feat,
                     gb, sx, sy, sl, ss, out_weight, out_bias, y);
  hipLaunchKernelGGL(finalize_kernel, dim3(1), dim3(32), 0, stream, entp, sqp,
                     y + (size_t)BB * NPTS);
}